// TAGConv_79207786873169
// MI455X (gfx1250) — compile-verified
//
#include <hip/hip_runtime.h>

#define DIN   128
#define DCAT  384
#define DOUT  128

typedef float v2f __attribute__((ext_vector_type(2)));
typedef float v8f __attribute__((ext_vector_type(8)));

__device__ __forceinline__ void atomic_add_f32(float* p, float v) {
    // lowers to global_atomic_add_f32 on gfx1250
    unsafeAtomicAdd(p, v);
}

// ---------------- utility: zero fill ----------------
__global__ void tag_zero_f32(float* __restrict__ p, long n) {
    long i = (long)blockIdx.x * blockDim.x + threadIdx.x;
    long stride = (long)gridDim.x * blockDim.x;
    for (; i < n; i += stride) p[i] = 0.0f;
}

// ---------------- degree via scatter-add of ones ----------------
__global__ void tag_degree(const int* __restrict__ dst, int E, float* __restrict__ deg) {
    int e = blockIdx.x * blockDim.x + threadIdx.x;
    if (e < E) atomic_add_f32(&deg[dst[e]], 1.0f);
}

// ---------------- norm = clip(deg,1)^-0.5 (in place) ----------------
__global__ void tag_norm(float* __restrict__ deg_norm, int N) {
    int i = blockIdx.x * blockDim.x + threadIdx.x;
    if (i < N) {
        float d = deg_norm[i];
        deg_norm[i] = 1.0f / sqrtf(fmaxf(d, 1.0f));
    }
}

// ---------------- one hop: agg[dst] += h[src] * norm[src] ----------------
// wave per edge: 32 lanes x float4 = 128 features
__global__ __launch_bounds__(256)
void tag_scatter(const float* __restrict__ h, const float* __restrict__ norm,
                 const int* __restrict__ src, const int* __restrict__ dst,
                 float* __restrict__ agg, int E) {
    long tid = (long)blockIdx.x * blockDim.x + threadIdx.x;
    int e    = (int)(tid >> 5);
    int lane = (int)(tid & 31);
    if (e >= E) return;
    int s = src[e];
    int d = dst[e];
    float ns = norm[s];
    const float4 v = *(const float4*)&h[(long)s * DIN + lane * 4];
    float* o = &agg[(long)d * DIN + lane * 4];
    atomic_add_f32(o + 0, v.x * ns);
    atomic_add_f32(o + 1, v.y * ns);
    atomic_add_f32(o + 2, v.z * ns);
    atomic_add_f32(o + 3, v.w * ns);
}

// ---------------- in-place post-scale: h[i,:] *= norm[i] ----------------
__global__ __launch_bounds__(256)
void tag_postscale(float* __restrict__ h, const float* __restrict__ norm, int N) {
    long tid = (long)blockIdx.x * blockDim.x + threadIdx.x;
    int i    = (int)(tid >> 5);
    int lane = (int)(tid & 31);
    if (i >= N) return;
    float nv = norm[i];
    float4* p = (float4*)&h[(long)i * DIN + lane * 4];
    float4 v = *p;
    v.x *= nv; v.y *= nv; v.z *= nv; v.w *= nv;
    *p = v;
}

// ---------------- GEMM: out[n,o] = sum_k cat[n,k] * W[o,k] + b[o] ----------------
// cat = [feat | h1 | h2] virtually, via three segment base pointers.
// One wave computes a 16x128 strip: 8 independent 16x16 accumulators so the
// 8 WMMA chains interleave in the matrix pipe, and each A fragment feeds 8 WMMAs.
__global__ __launch_bounds__(128)
void tag_gemm_wmma(const float* __restrict__ feat, const float* __restrict__ h1,
                   const float* __restrict__ h2, const float* __restrict__ W,
                   const float* __restrict__ bias, float* __restrict__ out,
                   int N, int mtiles) {
    const int lane = threadIdx.x & 31;
    const int mt   = blockIdx.x * 4 + (threadIdx.x >> 5);
    if (mt >= mtiles) return;                // uniform per wave

    const int m0  = mt * 16;
    const int l15 = lane & 15;
    const bool hi = lane >= 16;

    int arow = m0 + l15;                     // A-fragment node row (M = lane&15)
    if (arow >= N) arow = N - 1;             // clamp (no-op: 100000 % 16 == 0)

    const float* Arows[3] = { feat + (long)arow * DIN,
                              h1   + (long)arow * DIN,
                              h2   + (long)arow * DIN };

    v8f c[8] = {};

#pragma unroll
    for (int seg = 0; seg < 3; ++seg) {
        const float* __restrict__ Ar = Arows[seg];
        const float* __restrict__ Ws = W + seg * DIN;   // K-segment offset in W rows
#pragma unroll 4
        for (int k = 0; k < DIN; k += 4) {
            // A 16x4 fp32 fragment: lanes<16 hold K=k,k+1 ; lanes>=16 hold K=k+2,k+3
            // -> exact per-lane float2, no selects needed.
            v2f a = *(const v2f*)(Ar + k + (hi ? 2 : 0));
#pragma unroll
            for (int nt = 0; nt < 8; ++nt) {
                // B 4x16 fp32 fragment (B[k][n] = W[n][k]):
                // VGPR0: lanes<16 -> K=k, lanes>=16 -> K=k+1 ; VGPR1: K=k+2 / K=k+3
                float4 w4 = *(const float4*)(Ws + (long)(nt * 16 + l15) * DCAT + k);
                v2f b;
                b.x = hi ? w4.y : w4.x;
                b.y = hi ? w4.w : w4.z;
                c[nt] = __builtin_amdgcn_wmma_f32_16x16x4_f32(false, a, false, b,
                                                              (short)0, c[nt],
                                                              false, false);
            }
        }
    }

    // D layout: VGPR v -> M = m0 + v + (hi?8:0), N = nt*16 + (lane&15)
    const int rbase = m0 + (hi ? 8 : 0);
#pragma unroll
    for (int nt = 0; nt < 8; ++nt) {
        const int col = nt * 16 + l15;
        const float bv = bias[col];
#pragma unroll
        for (int v = 0; v < 8; ++v) {
            int row = rbase + v;
            if (row < N) out[(long)row * DOUT + col] = c[nt][v] + bv;
        }
    }
}

extern "C" void kernel_launch(void* const* d_in, const int* in_sizes, int n_in,
                              void* d_out, int out_size, void* d_ws, size_t ws_size,
                              hipStream_t stream) {
    const float* feat = (const float*)d_in[0];   // [N, 128]
    const float* W    = (const float*)d_in[1];   // [128, 384]
    const float* bias = (const float*)d_in[2];   // [128]
    const int*   src  = (const int*)d_in[3];     // [E]
    const int*   dst  = (const int*)d_in[4];     // [E]
    float*       out  = (float*)d_out;           // [N, 128]

    const int N = in_sizes[0] / DIN;
    const int E = in_sizes[3];

    // workspace: norm [N] | h1 [N*128] | h2 [N*128]  (~103 MB for N=100000)
    float* norm = (float*)d_ws;
    float* h1   = norm + N;
    float* h2   = h1 + (size_t)N * DIN;

    const int TB = 256;
    const long featN = (long)N * DIN;

    // degree -> norm
    tag_zero_f32<<<(N + TB - 1) / TB, TB, 0, stream>>>(norm, N);
    tag_degree<<<(E + TB - 1) / TB, TB, 0, stream>>>(dst, E, norm);
    tag_norm<<<(N + TB - 1) / TB, TB, 0, stream>>>(norm, N);

    const long scatThreads = (long)E * 32;
    const long rowThreads  = (long)N * 32;

    // hop 1: h1 = norm * scatter(feat * norm[src])
    tag_zero_f32<<<(int)((featN + TB - 1) / TB), TB, 0, stream>>>(h1, featN);
    tag_scatter<<<(int)((scatThreads + TB - 1) / TB), TB, 0, stream>>>(feat, norm, src, dst, h1, E);
    tag_postscale<<<(int)((rowThreads + TB - 1) / TB), TB, 0, stream>>>(h1, norm, N);

    // hop 2: h2 = norm * scatter(h1 * norm[src])
    tag_zero_f32<<<(int)((featN + TB - 1) / TB), TB, 0, stream>>>(h2, featN);
    tag_scatter<<<(int)((scatThreads + TB - 1) / TB), TB, 0, stream>>>(h1, norm, src, dst, h2, E);
    tag_postscale<<<(int)((rowThreads + TB - 1) / TB), TB, 0, stream>>>(h2, norm, N);

    // fused concat + GEMM + bias via fp32 WMMA (16x128 strip per wave)
    const int mtiles = (N + 15) / 16;
    tag_gemm_wmma<<<(mtiles + 3) / 4, 128, 0, stream>>>(feat, h1, h2, W, bias, out, N, mtiles);
}